// FISTA_Layer_36936718745824
// MI455X (gfx1250) — compile-verified
//
#include <hip/hip_runtime.h>

typedef float v2f __attribute__((ext_vector_type(2)));
typedef float v8f __attribute__((ext_vector_type(8)));

#define BSZ  16
#define NPAT 256
#define XDIM 128
#define UDIM 64
#define TT   32
#define PP   64   // 8*8 pixels per patch

// ---------------------------------------------------------------------------
// Fused kernel: per (b,p) block computes
//   X_pred[b,p,:, :] = A(128x128) @ Xprev(128x32)      (Xprev = [x0, X[:,0..30]])
//   recon            = C( 64x128) @ Xt   (128x32)      (Xt    = X[:,0..31])
// and scatters recon into the patch->image video layout.
// 8 waves per block; each 16x16 output tile = 32x V_WMMA_F32_16X16X4_F32.
// ---------------------------------------------------------------------------
__global__ __launch_bounds__(256) void fista_pred_recon(
    const float* __restrict__ X, const float* __restrict__ x0,
    const float* __restrict__ A, const float* __restrict__ C,
    float* __restrict__ vid, float* __restrict__ pred)
{
  // Xs[k][col]: col 0 = x0[k], col 1+t = X[b,p,k,t]; stride 33 avoids bank conflicts
  __shared__ float Xs[XDIM * 33];

  const int bp  = blockIdx.x;     // b*256 + p
  const int b   = bp >> 8;
  const int p   = bp & 255;
  const int tid = threadIdx.x;

  const float* Xbase  = X  + (size_t)bp * (XDIM * TT);
  const float* x0base = x0 + (size_t)bp * XDIM;

  for (int idx = tid; idx < XDIM * 33; idx += 256) {
    int k = idx / 33, col = idx - k * 33;
    Xs[idx] = (col == 0) ? x0base[k] : Xbase[k * TT + (col - 1)];
  }
  __syncthreads();

  const int wave = tid >> 5;
  const int lane = tid & 31;
  const int half = lane >> 4;   // K-pair select in A/B fragments
  const int lm   = lane & 15;   // M (A) / N (B,D) index within tile

  // ---------------- X_pred: wave w -> M-tile w, both N-tiles ----------------
  {
    const float* Arow = A + (wave * 16 + lm) * XDIM;
    #pragma unroll
    for (int nt = 0; nt < 2; ++nt) {
      const int ncol = nt * 16 + lm;            // Xprev col == Xs col (0..31)
      v8f acc = {};
      #pragma unroll 8
      for (int k0 = 0; k0 < XDIM; k0 += 4) {
        const int ka = k0 + 2 * half;
        v2f af; af.x = Arow[ka];            af.y = Arow[ka + 1];
        v2f bf; bf.x = Xs[ka * 33 + ncol];  bf.y = Xs[(ka + 1) * 33 + ncol];
        acc = __builtin_amdgcn_wmma_f32_16x16x4_f32(false, af, false, bf,
                                                    (short)0, acc, false, false);
      }
      // D layout: VGPR v holds row m = 16*wave + v + 8*half, col n = nt*16+lm
      float* outp = pred + (size_t)bp * (XDIM * TT) + (nt * 16 + lm);
      #pragma unroll
      for (int v = 0; v < 8; ++v) {
        const int m = wave * 16 + v + 8 * half;
        outp[(size_t)m * TT] = acc[v];
      }
    }
  }

  // ---------------- recon: 8 tiles; wave w -> mt = w&3, nt = w>>2 ----------
  {
    const int mt = wave & 3, nt = wave >> 2;
    const float* Crow = C + (mt * 16 + lm) * XDIM;
    const int ncol = nt * 16 + lm + 1;          // Xt col = Xs col + 1
    v8f acc = {};
    #pragma unroll 8
    for (int k0 = 0; k0 < XDIM; k0 += 4) {
      const int ka = k0 + 2 * half;
      v2f af; af.x = Crow[ka];            af.y = Crow[ka + 1];
      v2f bf; bf.x = Xs[ka * 33 + ncol];  bf.y = Xs[(ka + 1) * 33 + ncol];
      acc = __builtin_amdgcn_wmma_f32_16x16x4_f32(false, af, false, bf,
                                                  (short)0, acc, false, false);
    }
    // patch -> image scatter: p = gy*16+gx, j = py*8+px, h=gy*8+py, w=gx*8+px
    const int gy = p >> 4, gx = p & 15;
    const int t  = nt * 16 + lm;
    #pragma unroll
    for (int v = 0; v < 8; ++v) {
      const int j  = mt * 16 + v + 8 * half;
      const int py = j >> 3, px = j & 7;
      const int h  = gy * 8 + py, w = gx * 8 + px;
      vid[(((size_t)b * 128 + h) * 128 + w) * TT + t] = acc[v];
    }
  }
}

// ---------------------------------------------------------------------------
// X_U[b,xd,t] = 0.1*(1+exp(-(Bm@U)[xd]))/2 * sum_p |X[b,p,xd,t]|
// One wave per (b,xd): lane = t. X re-read is L2-resident (64MB << 192MB L2).
// ---------------------------------------------------------------------------
__global__ __launch_bounds__(256) void fista_xu(
    const float* __restrict__ X, const float* __restrict__ U,
    const float* __restrict__ Bm, float* __restrict__ xu)
{
  const int b  = blockIdx.x >> 4;
  const int xd = (blockIdx.x & 15) * 8 + (threadIdx.x >> 5);
  const int t  = threadIdx.x & 31;

  const float* xp = X + ((size_t)b * NPAT * XDIM + xd) * TT + t;
  float s = 0.f;
  #pragma unroll 4
  for (int p = 0; p < NPAT; ++p) s += fabsf(xp[(size_t)p * XDIM * TT]);

  const float* urow = U + (size_t)b * UDIM * TT + t;
  const float* brow = Bm + xd * UDIM;
  float dot = 0.f;
  #pragma unroll 8
  for (int u = 0; u < UDIM; ++u) dot += brow[u] * urow[(size_t)u * TT];

  xu[((size_t)b * XDIM + xd) * TT + t] = 0.05f * (1.f + expf(-dot)) * s;
}

// ---------------------------------------------------------------------------
extern "C" void kernel_launch(void* const* d_in, const int* in_sizes, int n_in,
                              void* d_out, int out_size, void* d_ws, size_t ws_size,
                              hipStream_t stream) {
  (void)in_sizes; (void)n_in; (void)out_size; (void)d_ws; (void)ws_size;
  const float* X  = (const float*)d_in[0];
  const float* U  = (const float*)d_in[1];
  const float* x0 = (const float*)d_in[2];
  const float* A  = (const float*)d_in[3];
  const float* Bm = (const float*)d_in[4];
  const float* C  = (const float*)d_in[5];

  float* out  = (float*)d_out;
  float* vid  = out;                                       // [16,1,128,128,32]
  float* pred = vid  + (size_t)BSZ * 128 * 128 * TT;       // [16,1,256,128,1,32]
  float* xu   = pred + (size_t)BSZ * NPAT * XDIM * TT;     // [16,1,1,128,1,32]

  fista_pred_recon<<<BSZ * NPAT, 256, 0, stream>>>(X, x0, A, C, vid, pred);
  fista_xu<<<BSZ * (XDIM / 8), 256, 0, stream>>>(X, U, Bm, xu);
}